// VectorQuantizer_16509854286430
// MI455X (gfx1250) — compile-verified
//
#include <hip/hip_runtime.h>

typedef __attribute__((ext_vector_type(2))) float v2f;
typedef __attribute__((ext_vector_type(8))) float v8f;

#define BATCH   16
#define SEQ     1024
#define DIM     256
#define NCODES  1024
#define MROWS   (BATCH * SEQ)   // 16384 flattened rows
#define KSTEPS  (DIM / 4)       // 64 WMMA k-steps (K=4 each)
#define NTILES  (NCODES / 16)   // 64 code tiles
#define WAVES_PER_BLOCK 4
#define THREADS (WAVES_PER_BLOCK * 32)
#define BROW    260             // padded LDS row stride (floats): bank-conflict-free

// ---------------------------------------------------------------------------
// Kernel 1: csq[n] = -0.5 * ||codebook[n]||^2   (one wave32 per code)
// ---------------------------------------------------------------------------
__global__ void vq_csq_kernel(const float* __restrict__ codebook,
                              float* __restrict__ csq) {
  const int wave = (blockIdx.x * blockDim.x + threadIdx.x) >> 5;
  const int lane = threadIdx.x & 31;
  if (wave >= NCODES) return;
  const float4* r4 = (const float4*)(codebook + wave * DIM);
  float4 x = r4[lane];
  float4 y = r4[lane + 32];
  float s = x.x * x.x + x.y * x.y + x.z * x.z + x.w * x.w +
            y.x * y.x + y.y * y.y + y.z * y.z + y.w * y.w;
#pragma unroll
  for (int off = 16; off >= 1; off >>= 1) s += __shfl_xor(s, off, 32);
  if (lane == 0) csq[wave] = -0.5f * s;
}

// ---------------------------------------------------------------------------
// Kernel 2: fused GEMM + argmax + gather.
// 4 waves share a double-buffered LDS codebook tile filled by async DMA
// (global_load_async_to_lds_b128); each wave owns 16 rows of z_e resident in
// VGPRs and chains 64 V_WMMA_F32_16X16X4_F32 per tile reading B from LDS.
// ---------------------------------------------------------------------------
__global__ void __launch_bounds__(THREADS, 1)
vq_main_kernel(const float* __restrict__ z_e,
               const float* __restrict__ codebook,
               const float* __restrict__ csq_nh,   // -0.5*||c||^2
               float* __restrict__ z_q,
               float* __restrict__ idx_out) {
  __shared__ float tileB[2][16 * BROW];            // 2 x 16.25 KB, padded rows

  const int lane = threadIdx.x & 31;
  const int wave = threadIdx.x >> 5;
  const int tile = blockIdx.x * WAVES_PER_BLOCK + wave;  // 0..1023
  const int m0   = tile * 16;

  // Fragment layout (wave32):
  //   A 16x4 f32: row = lane&15, k-pair offset = (lane>>4)*2
  //   B 4x16 f32: col = lane&15, k-pair offset = (lane>>4)*2
  const int mrow = lane & 15;
  const int koff = (lane >> 4) * 2;

  // Async DMA fill of one 16x256 codebook tile into LDS (padded stride BROW).
  // 128 threads x 8 chunks x 16B = 16 KB; tracked on ASYNCcnt.
  auto fill_tile = [&](int t, int buf) {
    const float* g = codebook + (size_t)t * 16 * DIM;   // tile is contiguous
    const unsigned lbase = (unsigned)(uintptr_t)(&tileB[buf][0]);
    const int tid = threadIdx.x;
#pragma unroll
    for (int p = 0; p < 8; ++p) {
      const int chunk = tid + p * THREADS;   // 0..1023 16B-chunks
      const int row = chunk >> 6;            // 64 chunks per codebook row
      const int c4  = (chunk & 63) * 4;      // float offset within row
      const float*  gsrc  = g + row * DIM + c4;
      const unsigned laddr = lbase + (unsigned)((row * BROW + c4) * 4);
      asm volatile("global_load_async_to_lds_b128 %0, %1, off"
                   :: "v"(laddr), "v"(gsrc) : "memory");
    }
  };

  // Resident A tile: 16x256 across 64 k-step fragments (128 VGPRs).
  v2f afrag[KSTEPS];
  const float* abase = z_e + (m0 + mrow) * DIM + koff;
#pragma unroll
  for (int i = 0; i < KSTEPS; ++i)
    afrag[i] = *(const v2f*)(abase + 4 * i);

  float bestv[8];
  int   besti[8];
#pragma unroll
  for (int v = 0; v < 8; ++v) { bestv[v] = -3.4e38f; besti[v] = 0; }

  fill_tile(0, 0);                                  // prologue

  for (int t = 0; t < NTILES; ++t) {
    const int cur  = t & 1;
    const int code = t * 16 + mrow;                 // this lane's code column

    asm volatile("s_wait_asynccnt 0x0" ::: "memory");  // our buffer is filled
    __syncthreads();                                    // ...for all waves

    if (t + 1 < NTILES) fill_tile(t + 1, 1 - cur);  // overlap next fill

    // acc = -0.5*||c_n||^2 ; WMMA adds z.c -> argmax(acc) == argmin distance
    const float cinit = csq_nh[code];
    v8f acc;
#pragma unroll
    for (int v = 0; v < 8; ++v) acc[v] = cinit;

    // Bank-conflict-free: dword bank = (4*mrow + 4i + koff) % 64 covers all
    // 64 banks exactly once per ds_load_b64 across the wave.
    const float* bb = &tileB[cur][mrow * BROW + koff];
#pragma unroll
    for (int i = 0; i < KSTEPS; ++i) {
      v2f b = *(const v2f*)(bb + 4 * i);
      acc = __builtin_amdgcn_wmma_f32_16x16x4_f32(
          /*neg_a=*/false, afrag[i], /*neg_b=*/false, b,
          /*c_mod=*/(short)0, acc, /*reuse_a=*/false, /*reuse_b=*/false);
    }

    // Running per-(row,lane) argmax; strict > keeps lowest code on ties.
#pragma unroll
    for (int v = 0; v < 8; ++v) {
      const float d = acc[v];
      if (d > bestv[v]) { bestv[v] = d; besti[v] = code; }
    }
  }

  // Butterfly argmax across the 16 lanes of each half-wave (N dimension).
#pragma unroll
  for (int v = 0; v < 8; ++v) {
    float bv = bestv[v];
    int   bi = besti[v];
#pragma unroll
    for (int off = 8; off >= 1; off >>= 1) {
      const float ov = __shfl_xor(bv, off, 32);
      const int   oi = __shfl_xor(bi, off, 32);
      if (ov > bv || (ov == bv && oi < bi)) { bv = ov; bi = oi; }
    }
    bestv[v] = bv;
    besti[v] = bi;
  }

  // Rows m0+v live in lanes 0..15 (slot v); rows m0+8+v in lanes 16..31.
  if (lane == 0 || lane == 16) {
    const int rbase = m0 + (lane >> 4) * 8;
#pragma unroll
    for (int v = 0; v < 8; ++v)
      idx_out[rbase + v] = (float)besti[v];
  }

  // Gather z_q rows: whole wave copies each winning codebook row (coalesced).
#pragma unroll
  for (int m = 0; m < 16; ++m) {
    const int bi = __shfl(besti[m & 7], (m < 8) ? 0 : 16, 32);
    const float4* crow = (const float4*)(codebook + bi * DIM);
    float4* orow = (float4*)(z_q + (m0 + m) * DIM);
    orow[lane]      = crow[lane];
    orow[lane + 32] = crow[lane + 32];
  }
}

// ---------------------------------------------------------------------------
extern "C" void kernel_launch(void* const* d_in, const int* in_sizes, int n_in,
                              void* d_out, int out_size, void* d_ws, size_t ws_size,
                              hipStream_t stream) {
  const float* z_e      = (const float*)d_in[0];  // [16,1024,256] f32
  const float* codebook = (const float*)d_in[1];  // [1024,256] f32
  float* z_q     = (float*)d_out;                  // [16,1024,256]
  float* idx_out = (float*)d_out + (size_t)MROWS * DIM;  // [16,1024] as float
  float* csq     = (float*)d_ws;                   // 1024 floats scratch

  vq_csq_kernel<<<(NCODES * 32) / 256, 256, 0, stream>>>(codebook, csq);

  vq_main_kernel<<<MROWS / 16 / WAVES_PER_BLOCK, THREADS, 0, stream>>>(
      z_e, codebook, csq, z_q, idx_out);
}